// MultiHeadAttentionQuantum_22582938042546
// MI455X (gfx1250) — compile-verified
//
#include <hip/hip_runtime.h>
#include <math.h>

#define NQ     8
#define SEQ    1024
#define BATCH  32
#define NHEAD  2
#define DK     4
#define NTOK   (BATCH * SEQ)

typedef __attribute__((ext_vector_type(2))) float v2f;
typedef __attribute__((ext_vector_type(8))) float v8f;

#if defined(__has_builtin)
#if __has_builtin(__builtin_amdgcn_wmma_f32_16x16x4_f32)
#define HAVE_WMMA_F32X4 1
#endif
#endif

__device__ __forceinline__ float lane_xor(float v, int mask) {
  return __shfl_xor(v, mask, 32);
}

// ---------------------------------------------------------------------------
// 8-qubit circuit simulator: one token per wave32.
// Amplitude index j (8 bits): qubit q <-> bit (7-q).  j = (lane<<3) | r.
// Lane bits (j[7:3]) = qubits 0..4 -> cross-lane shfl_xor; reg bits (j[2:0])
// = qubits 5..7 -> compile-time register permutation (fully unrolled).
// ---------------------------------------------------------------------------
__global__ __launch_bounds__(256)
void qcircuit_kernel(const float* __restrict__ x, const float* __restrict__ w,
                     float* __restrict__ out, int ntok) {
  const int gid  = blockIdx.x * blockDim.x + threadIdx.x;
  const int wave = gid >> 5;
  const int lane = threadIdx.x & 31;
  if (wave >= ntok) return;

  // per-token embedding angles
  float ca[NQ], sa[NQ];
  const float* xt = x + (size_t)wave * NQ;
#pragma unroll
  for (int q = 0; q < NQ; ++q) {
    float h = 0.5f * xt[q];
    ca[q] = __cosf(h);
    sa[q] = __sinf(h);
  }
  // layer weights (shared by all tokens; L2-resident)
  float cw[2][NQ], sw[2][NQ];
#pragma unroll
  for (int l = 0; l < 2; ++l)
#pragma unroll
    for (int q = 0; q < NQ; ++q) {
      float h = 0.5f * w[l * NQ + q];
      cw[l][q] = __cosf(h);
      sw[l][q] = __sinf(h);
    }

  // product state: amp(j) = (-i)^popcount(j) * prod_q (bit ? sin : cos)
  float re[8], im[8];
#pragma unroll
  for (int r = 0; r < 8; ++r) {
    int j = (lane << 3) | r;
    float m = 1.0f;
#pragma unroll
    for (int q = 0; q < NQ; ++q)
      m *= ((j >> (7 - q)) & 1) ? sa[q] : ca[q];
    int k = __popc(j) & 3;
    re[r] = (k == 0) ? m : ((k == 2) ? -m : 0.0f);
    im[r] = (k == 1) ? -m : ((k == 3) ? m : 0.0f);
  }

#pragma unroll
  for (int l = 0; l < 2; ++l) {
    // RX on every qubit: new = c*a(j) - i*s*a(j^mask)
#pragma unroll
    for (int q = 0; q < NQ; ++q) {
      const int mask = 1 << (7 - q);
      const float c = cw[l][q], s = sw[l][q];
      float pre[8], pim[8];
#pragma unroll
      for (int r = 0; r < 8; ++r) {
        int rp = r ^ (mask & 7);
        pre[r] = re[rp];
        pim[r] = im[rp];
      }
      if (mask >> 3) {
#pragma unroll
        for (int r = 0; r < 8; ++r) {
          pre[r] = lane_xor(pre[r], mask >> 3);
          pim[r] = lane_xor(pim[r], mask >> 3);
        }
      }
#pragma unroll
      for (int r = 0; r < 8; ++r) {
        float nre = c * re[r] + s * pim[r];
        float nim = c * im[r] - s * pre[r];
        re[r] = nre;
        im[r] = nim;
      }
    }
    // CNOT ring: ctrl q -> tgt (q+1)%8; amp(j) <- amp(j^tmask) when ctrl bit set
#pragma unroll
    for (int q = 0; q < NQ; ++q) {
      const int tgt   = (q + 1) & 7;
      const int cmask = 1 << (7 - q);
      const int tmask = 1 << (7 - tgt);
      float pre[8], pim[8];
#pragma unroll
      for (int r = 0; r < 8; ++r) {
        int rp = r ^ (tmask & 7);
        pre[r] = re[rp];
        pim[r] = im[rp];
      }
      if (tmask >> 3) {
#pragma unroll
        for (int r = 0; r < 8; ++r) {
          pre[r] = lane_xor(pre[r], tmask >> 3);
          pim[r] = lane_xor(pim[r], tmask >> 3);
        }
      }
#pragma unroll
      for (int r = 0; r < 8; ++r) {
        int j = (lane << 3) | r;
        bool cs = (j & cmask) != 0;
        re[r] = cs ? pre[r] : re[r];
        im[r] = cs ? pim[r] : im[r];
      }
    }
  }

  // probabilities and <Z_q>
  float p[8];
#pragma unroll
  for (int r = 0; r < 8; ++r) p[r] = re[r] * re[r] + im[r] * im[r];

  float ez[NQ];
#pragma unroll
  for (int q = 0; q < NQ; ++q) {
    const int pbit = 7 - q;
    float acc = 0.0f;
#pragma unroll
    for (int r = 0; r < 8; ++r) {
      int j = (lane << 3) | r;
      acc += ((j >> pbit) & 1) ? -p[r] : p[r];
    }
    ez[q] = acc;
  }
#pragma unroll
  for (int off = 16; off > 0; off >>= 1)
#pragma unroll
    for (int q = 0; q < NQ; ++q)
      ez[q] += lane_xor(ez[q], off);

  if (lane == 0) {
    float4* o = (float4*)(out + (size_t)wave * NQ);
    o[0] = make_float4(ez[0], ez[1], ez[2], ez[3]);
    o[1] = make_float4(ez[4], ez[5], ez[6], ez[7]);
  }
}

// ---------------------------------------------------------------------------
// Attention: block = (b, h, 16-query tile).  4 waves; each wave covers 16 key
// tiles with V_WMMA_F32_16X16X4_F32 (exact d_k=4 fit).
//
// Scores are bounded (|Q.K/2| <= 2 since entries are <Z> in [-1,1]), so the
// softmax needs no max-subtraction: accumulate sum(e) and sum(e*V) directly
// per lane across all key tiles; one shfl_xor reduction AFTER the loop.
// The K/V tile loads are double-buffered so the WMMA never waits on VMEM.
//
// A/B layout (32-bit, 16x4): lanes 0-15 carry K={0,1}, lanes 16-31 K={2,3}.
// C layout: VGPR r -> row r (+8 for upper lane half), lane&15 -> column.
// ---------------------------------------------------------------------------
__global__ __launch_bounds__(128)
void attn_kernel(const float* __restrict__ Q, const float* __restrict__ K,
                 const float* __restrict__ V, float* __restrict__ ctx) {
  const int blk  = blockIdx.x;
  const int qt   = blk & 63;
  const int h    = (blk >> 6) & 1;
  const int b    = blk >> 7;
  const int tid  = threadIdx.x;
  const int wave = tid >> 5;
  const int lane = tid & 31;
  const int lh   = lane & 15;  // position inside 16-lane half
  const int hi   = lane >> 4;  // half id

  const int tokbase = b * SEQ;
  const int cof     = h * DK;

  // A tile: 16x4 slab of Q
  const float* qrow = Q + (size_t)(tokbase + qt * 16 + lh) * NQ + cof + hi * 2;
  v2f A;
  A.x = qrow[0];
  A.y = qrow[1];

  // per-lane accumulators: rows hi*8 .. hi*8+7 of the 16-row tile,
  // this lane's key column only (reduced across the half after the loop)
  float sum[8], acc0[8], acc1[8], acc2[8], acc3[8];
#pragma unroll
  for (int r = 0; r < 8; ++r) {
    sum[r] = 0.0f;
    acc0[r] = acc1[r] = acc2[r] = acc3[r] = 0.0f;
  }

  // double-buffered K row (v2f) and V row (float4)
  const float* krow0 =
      K + (size_t)(tokbase + wave * 16 + lh) * NQ + cof + hi * 2;
  const float4* vrow0 =
      (const float4*)(V + (size_t)(tokbase + wave * 16 + lh) * NQ + cof);
  v2f Bv;
  Bv.x = krow0[0];
  Bv.y = krow0[1];
  float4 Vv = vrow0[0];

  for (int kt = wave; kt < SEQ / 16; kt += 4) {
    // issue next tile's loads early (clamped reload of first tile at the end:
    // harmless, keeps the loop branch-free)
    const int ktn = (kt + 4 < SEQ / 16) ? (kt + 4) : wave;
    const float* krown =
        K + (size_t)(tokbase + ktn * 16 + lh) * NQ + cof + hi * 2;
    const float4* vrown =
        (const float4*)(V + (size_t)(tokbase + ktn * 16 + lh) * NQ + cof);
    v2f Bn;
    Bn.x = krown[0];
    Bn.y = krown[1];
    float4 Vn = vrown[0];

    v8f C = {0.f, 0.f, 0.f, 0.f, 0.f, 0.f, 0.f, 0.f};
#ifdef HAVE_WMMA_F32X4
    C = __builtin_amdgcn_wmma_f32_16x16x4_f32(false, A, false, Bv, (short)0, C,
                                              false, false);
#else
#pragma unroll
    for (int r = 0; r < 8; ++r) {
      const float* qr = Q + (size_t)(tokbase + qt * 16 + hi * 8 + r) * NQ + cof;
      const float* kr = K + (size_t)(tokbase + kt * 16 + lh) * NQ + cof;
      C[r] = qr[0] * kr[0] + qr[1] * kr[1] + qr[2] * kr[2] + qr[3] * kr[3];
    }
#endif

#pragma unroll
    for (int r = 0; r < 8; ++r) {
      float e = __expf(C[r] * 0.5f);  // 1/sqrt(d_k); |arg| <= 2, no overflow
      sum[r]  += e;
      acc0[r] += e * Vv.x;
      acc1[r] += e * Vv.y;
      acc2[r] += e * Vv.z;
      acc3[r] += e * Vv.w;
    }

    Bv = Bn;
    Vv = Vn;
  }

  // one reduction across the 16-lane half (key columns), after the loop
#pragma unroll
  for (int m2 = 1; m2 <= 8; m2 <<= 1) {
#pragma unroll
    for (int r = 0; r < 8; ++r) {
      sum[r]  += lane_xor(sum[r], m2);
      acc0[r] += lane_xor(acc0[r], m2);
      acc1[r] += lane_xor(acc1[r], m2);
      acc2[r] += lane_xor(acc2[r], m2);
      acc3[r] += lane_xor(acc3[r], m2);
    }
  }

  // merge the 4 wave-partials (pure sums; no max bookkeeping needed)
  __shared__ float smem[4][16][5];
  if (lh == 0) {
#pragma unroll
    for (int r = 0; r < 8; ++r) {
      int row = hi * 8 + r;
      smem[wave][row][0] = sum[r];
      smem[wave][row][1] = acc0[r];
      smem[wave][row][2] = acc1[r];
      smem[wave][row][3] = acc2[r];
      smem[wave][row][4] = acc3[r];
    }
  }
  __syncthreads();
  if (tid < 16) {
    int row = tid;
    float S = 0.f, A0 = 0.f, A1 = 0.f, A2 = 0.f, A3 = 0.f;
#pragma unroll
    for (int w2 = 0; w2 < 4; ++w2) {
      S  += smem[w2][row][0];
      A0 += smem[w2][row][1];
      A1 += smem[w2][row][2];
      A2 += smem[w2][row][3];
      A3 += smem[w2][row][4];
    }
    float inv = 1.0f / S;
    float* o = ctx + (size_t)(tokbase + qt * 16 + row) * NQ + cof;
    o[0] = A0 * inv;
    o[1] = A1 * inv;
    o[2] = A2 * inv;
    o[3] = A3 * inv;
  }
}

// ---------------------------------------------------------------------------
extern "C" void kernel_launch(void* const* d_in, const int* in_sizes, int n_in,
                              void* d_out, int out_size, void* d_ws,
                              size_t ws_size, hipStream_t stream) {
  const float* x  = (const float*)d_in[0];
  const float* wq = (const float*)d_in[1];
  const float* wk = (const float*)d_in[2];
  const float* wv = (const float*)d_in[3];
  const float* wc = (const float*)d_in[4];
  float* out = (float*)d_out;

  float* Qb = (float*)d_ws;                 // [NTOK, 8]
  float* Kb = Qb + (size_t)NTOK * NQ;       // [NTOK, 8]
  float* Vb = Kb + (size_t)NTOK * NQ;       // [NTOK, 8]
  float* Cb = Vb + (size_t)NTOK * NQ;       // [NTOK, 8] (attention context)

  const int circ_blocks = (NTOK + 7) / 8;   // 8 tokens (waves) per 256-thread block
  qcircuit_kernel<<<circ_blocks, 256, 0, stream>>>(x, wq, Qb, NTOK);
  qcircuit_kernel<<<circ_blocks, 256, 0, stream>>>(x, wk, Kb, NTOK);
  qcircuit_kernel<<<circ_blocks, 256, 0, stream>>>(x, wv, Vb, NTOK);
  attn_kernel<<<BATCH * NHEAD * (SEQ / 16), 128, 0, stream>>>(Qb, Kb, Vb, Cb);
  qcircuit_kernel<<<circ_blocks, 256, 0, stream>>>(Cb, wc, out, NTOK);
}